// GroupVectorQuantizer_56581899157656
// MI455X (gfx1250) — compile-verified
//
#include <hip/hip_runtime.h>

typedef __attribute__((ext_vector_type(2))) float v2f;
typedef __attribute__((ext_vector_type(8))) float v8f;

#define CB_SIZE   1024
#define C_DIM     256
#define B_DIM     32
#define H_DIM     32
#define W_DIM     32
#define HW        (H_DIM * W_DIM)          // 1024
#define N_TOK     (B_DIM * HW)             // 32768
#define CODES_PER_GROUP 128
#define ZQ_ELEMS  (B_DIM * C_DIM * HW)     // 8388608
#define NUM_WAVES_B 2048                   // 32 batches * 8 groups * 8 token-tiles

// ---------------------------------------------------------------------------
// Kernel 1: codebook[m][n] = sum_k emb[m][k] * proj[n][k]   (emb @ proj.T)
// One wave computes one 16x16 tile via V_WMMA_F32_16X16X4_F32, K = 256.
// A layout (ISA 7.12.2, 32-bit A 16x4): lane row = l&15; VGPR j holds
// K = 2*(l>>4) + j. B mirrors A (rows striped across lanes).
// ---------------------------------------------------------------------------
__global__ __launch_bounds__(32) void codebook_gemm(
    const float* __restrict__ emb, const float* __restrict__ proj,
    float* __restrict__ cb) {
  const int tile = blockIdx.x;            // 64 M-tiles x 16 N-tiles
  const int m0 = (tile >> 4) << 4;
  const int n0 = (tile & 15) << 4;
  const int l  = threadIdx.x;
  const int lx = l & 15;
  const int hi = l >> 4;

  const float* arow = emb  + (size_t)(m0 + lx) * C_DIM;
  const float* brow = proj + (size_t)(n0 + lx) * C_DIM;

  v8f acc = {};
#pragma unroll 4
  for (int kk = 0; kk < C_DIM; kk += 4) {
    const int k = kk + 2 * hi;                  // consecutive K pair per lane
    v2f a = *(const v2f*)(arow + k);
    v2f b = *(const v2f*)(brow + k);
    acc = __builtin_amdgcn_wmma_f32_16x16x4_f32(
        false, a, false, b, (short)0, acc, false, false);
  }

  float* out = cb + (size_t)n0 + lx;
#pragma unroll
  for (int r = 0; r < 8; ++r)
    out[(size_t)(m0 + r + 8 * hi) * C_DIM] = acc[r];
}

// ---------------------------------------------------------------------------
// Kernel 2: per-code squared norms
// ---------------------------------------------------------------------------
__global__ void code_norms(const float* __restrict__ cb,
                           float* __restrict__ cnorm) {
  const int i = blockIdx.x * blockDim.x + threadIdx.x;
  if (i < CB_SIZE) {
    const float* row = cb + (size_t)i * C_DIM;
    float s = 0.f;
#pragma unroll 4
    for (int c = 0; c < C_DIM; ++c) { float v = row[c]; s += v * v; }
    cnorm[i] = s;
  }
}

// ---------------------------------------------------------------------------
// Kernel 3: per-wave argmin: 16 tokens of one (batch, group) vs the group's
// 128 codes. Scores via 64 k-steps x 8 WMMA_F32_16X16X4_F32 (A reused across
// the 8 code tiles). dist = ||c||^2 - 2*score (token norm constant per row).
// Per-row argmin via in-lane min over the 8 code tiles + width-16 shuffle
// tree; lane lx of each half keeps the row-(lx+8*hi) result in a register,
// then writes idx (float to d_out, int to ws for the scatter kernel).
// ---------------------------------------------------------------------------
__global__ __launch_bounds__(32) void vq_argmin(
    const float* __restrict__ z, const float* __restrict__ cb,
    const float* __restrict__ cnorm, float* __restrict__ out,
    int* __restrict__ ws_idx) {
  const int wid = blockIdx.x;              // 0..2047
  const int b   = wid >> 6;                // batch
  const int rem = wid & 63;
  const int g   = rem >> 3;                // group = w>>2
  const int t8  = rem & 7;                 // token tile within (b,g)
  const int l   = threadIdx.x;
  const int lx  = l & 15;
  const int hi  = l >> 4;

  // A-fragment row (token m = lx): token t in group -> (h, w)
  const int tA = t8 * 16 + lx;
  const int hA = tA >> 2;
  const int wA = g * 4 + (tA & 3);
  const float* zrow = z + (size_t)b * C_DIM * HW + hA * W_DIM + wA; // +k*HW

  const int code0 = g * CODES_PER_GROUP;
  const float* cbase = cb + (size_t)(code0 + lx) * C_DIM;

  v8f acc[8] = {};
#pragma unroll 1
  for (int kk = 0; kk < C_DIM; kk += 4) {
    const int k = kk + 2 * hi;
    v2f a;
    a.x = zrow[(size_t)k * HW];            // C-stride = H*W in z's layout
    a.y = zrow[(size_t)(k + 1) * HW];
#pragma unroll
    for (int ct = 0; ct < 8; ++ct) {
      v2f bf = *(const v2f*)(cbase + (size_t)ct * 16 * C_DIM + k);
      acc[ct] = __builtin_amdgcn_wmma_f32_16x16x4_f32(
          false, a, false, bf, (short)0, acc[ct], false, false);
    }
  }

  // Per-lane code-column norms for the 8 code tiles
  float cn[8];
#pragma unroll
  for (int ct = 0; ct < 8; ++ct) cn[ct] = cnorm[code0 + ct * 16 + lx];

  // Argmin per token row: D row M = r + 8*hi, col N = lx (+ 16*ct).
  int keep = 0;
#pragma unroll
  for (int r = 0; r < 8; ++r) {
    float bestv = 3.402823e38f;
    int   bestc = 0x7fffffff;
#pragma unroll
    for (int ct = 0; ct < 8; ++ct) {
      const float d = cn[ct] - 2.0f * acc[ct][r];
      const int   c = code0 + ct * 16 + lx;
      if (d < bestv || (d == bestv && c < bestc)) { bestv = d; bestc = c; }
    }
    // min across the 16 lanes of each half (jnp.argmin tie -> lowest index)
#pragma unroll
    for (int m = 8; m >= 1; m >>= 1) {
      const float ov = __shfl_xor(bestv, m, 16);
      const int   oc = __shfl_xor(bestc, m, 16);
      if (ov < bestv || (ov == bestv && oc < bestc)) { bestv = ov; bestc = oc; }
    }
    if (lx == r) keep = bestc;             // lane lx owns token lx + 8*hi
  }

  const size_t IDX_OFF = (size_t)ZQ_ELEMS + 3;
  if (lx < 8) {
    const int m = lx + 8 * hi;
    const int t = t8 * 16 + m;
    const int h = t >> 2;
    const int w = g * 4 + (t & 3);
    const int n = b * HW + h * W_DIM + w;
    out[IDX_OFF + n] = (float)keep;        // idx output (fp32 d_out)
    ws_idx[n] = keep;                      // int copy for scatter kernel
  }
}

// ---------------------------------------------------------------------------
// Kernel 4: coalesced zq scatter + loss partials. One block per (b, h): a
// full 32-wide W row spans all 8 groups, so each 32-lane wave stores/loads
// contiguous 128B lines of the [B,C,H,W] output. Codebook gathers hit L2.
// ---------------------------------------------------------------------------
__global__ __launch_bounds__(256) void vq_scatter(
    const float* __restrict__ z, const float* __restrict__ cb,
    const int* __restrict__ ws_idx, float* __restrict__ out,
    float* __restrict__ partial) {
  const int bx = blockIdx.x;               // 0..1023
  const int b  = bx >> 5;
  const int h  = bx & 31;
  const int t  = threadIdx.x;

  __shared__ int s_code[W_DIM];
  if (t < W_DIM) s_code[t] = ws_idx[b * HW + h * W_DIM + t];
  __syncthreads();

  const int w  = t & 31;                   // contiguous within each wave
  const int dc = t >> 5;                   // 8 C-planes in flight per block
  const float* crow = cb + (size_t)s_code[w] * C_DIM;

  float sq = 0.f;
#pragma unroll 1
  for (int c0 = 0; c0 < C_DIM; c0 += 8) {
    const int c = c0 + dc;
    const float v = crow[c];
    const size_t off = ((size_t)(b * C_DIM + c)) * HW + h * W_DIM + w;
    const float dz = v - z[off];
    out[off] = v;                          // zq_ste == zq numerically
    sq += dz * dz;
  }

  // deterministic block tree reduction
  __shared__ float s_red[256];
  s_red[t] = sq;
  __syncthreads();
#pragma unroll
  for (int s = 128; s > 0; s >>= 1) {
    if (t < s) s_red[t] += s_red[t + s];
    __syncthreads();
  }
  if (t == 0) partial[bx] = s_red[0];
}

// ---------------------------------------------------------------------------
// Kernel 5: deterministic fixed-order reduction of partials -> scalar losses
// ---------------------------------------------------------------------------
__global__ __launch_bounds__(32) void vq_finalize(
    const float* __restrict__ partial, float* __restrict__ out) {
  const int l = threadIdx.x;
  float s = 0.f;
#pragma unroll 1
  for (int k = 0; k < 1024 / 32; ++k) s += partial[l + 32 * k];
#pragma unroll
  for (int m = 16; m >= 1; m >>= 1) s += __shfl_xor(s, m, 32);
  if (l == 0) {
    const float msd = s / (float)ZQ_ELEMS; // mean((zq - zc)^2)
    out[ZQ_ELEMS + 0] = 1.25f * msd;       // loss
    out[ZQ_ELEMS + 1] = 0.25f * msd;       // commitment_loss
    out[ZQ_ELEMS + 2] = msd;               // codebook_loss
  }
}

// ---------------------------------------------------------------------------
extern "C" void kernel_launch(void* const* d_in, const int* in_sizes, int n_in,
                              void* d_out, int out_size, void* d_ws,
                              size_t ws_size, hipStream_t stream) {
  const float* z    = (const float*)d_in[0];
  const float* emb  = (const float*)d_in[1];
  const float* proj = (const float*)d_in[2];
  float* out = (float*)d_out;

  float* cb      = (float*)d_ws;                  // 1024*256 floats (1 MB)
  float* cnorm   = cb + (size_t)CB_SIZE * C_DIM;  // 1024 floats
  int*   ws_idx  = (int*)(cnorm + CB_SIZE);       // 32768 ints
  float* partial = (float*)(ws_idx + N_TOK);      // 1024 floats

  codebook_gemm<<<1024, 32, 0, stream>>>(emb, proj, cb);
  code_norms<<<4, 256, 0, stream>>>(cb, cnorm);
  vq_argmin<<<NUM_WAVES_B, 32, 0, stream>>>(z, cb, cnorm, out, ws_idx);
  vq_scatter<<<1024, 256, 0, stream>>>(z, cb, ws_idx, out, partial);
  vq_finalize<<<1, 32, 0, stream>>>(partial, out);
}